// GNN_17712445128828
// MI455X (gfx1250) — compile-verified
//
#include <hip/hip_runtime.h>

// ---------------------------------------------------------------------------
// GNN forward for MI455X (gfx1250, wave32).
// Dense math: V_WMMA_F32_16X16X4_F32 (fp32 kept: workload is gather/scatter +
// bandwidth bound; node tables fit in the 192MB L2 so gathers run at L2 BW).
// Edge-MLP A operand gathered on the fly ([h[src] | h[dst] | ea]) inside the
// GEMM with branch-free address selects; BN batch stats fused into the GEMM
// epilogue; main K-loop is completely unpredicated (bounds handled by row
// clamping + a single zero-padded tail step).
// ---------------------------------------------------------------------------

typedef float v2f __attribute__((ext_vector_type(2)));
typedef float v8f __attribute__((ext_vector_type(8)));

#define BN_EPS   1e-5f
#define NNODES   50000
#define NEDGES   800000
#define NGRAPHS  1024
#define MAXK     320
#define GWAVES   8
#define GTHREADS (GWAVES * 32)

// ---------------- utility kernels ----------------
__global__ void k_zero(float* __restrict__ p, long long n) {
  long long i  = (long long)blockIdx.x * blockDim.x + threadIdx.x;
  long long st = (long long)gridDim.x * blockDim.x;
  for (; i < n; i += st) p[i] = 0.f;
}

__global__ void k_bn_finalize(const float* __restrict__ cs, const float* __restrict__ css,
                              const float* __restrict__ gamma, const float* __restrict__ beta,
                              int N, float invM,
                              float* __restrict__ scale, float* __restrict__ shift) {
  int c = blockIdx.x * blockDim.x + threadIdx.x;
  if (c < N) {
    float m = cs[c] * invM;
    float v = css[c] * invM - m * m;            // biased variance (ddof=0)
    float s = gamma[c] * rsqrtf(fmaxf(v, 0.f) + BN_EPS);
    scale[c] = s;
    shift[c] = beta[c] - m * s;
  }
}

// act: 0 = identity, 1 = relu, 2 = tanh.  scale==nullptr -> no BN affine.
__global__ void k_apply(float* __restrict__ Y, long long total, int N,
                        const float* __restrict__ scale, const float* __restrict__ shift,
                        int act) {
  long long i  = (long long)blockIdx.x * blockDim.x + threadIdx.x;
  long long st = (long long)gridDim.x * blockDim.x;
  if (i >= total) return;
  int c     = (int)(i % N);          // one division per thread,
  int cstep = (int)(st % N);         // then incremental wraparound
  for (; i < total; i += st) {
    float v = Y[i];
    if (scale) v = v * scale[c] + shift[c];
    if (act == 1)      v = fmaxf(v, 0.f);
    else if (act == 2) v = tanhf(v);
    Y[i] = v;
    c += cstep; if (c >= N) c -= N;
  }
}

// ---------------- WMMA GEMM (Y = act-later(X*W + b), fused col stats) -------
// A-operand loader policies. init() clamps the row so every address is valid;
// rows m >= M only feed C rows that are never stored / never counted.
struct APlain {
  const float* X; int ldx;
  const float* row;
  __device__ void init(int m, int M) {
    int mm = (m < M) ? m : (M - 1);
    row = X + (long long)mm * ldx;
  }
  __device__ float get(int k) const { return row[k]; }                 // k < K
  __device__ float getz(int k, int K) const { return (k < K) ? row[k] : 0.f; }
};

struct AEdge {  // A row e = [ H[src[e]][0:ni] | H[dst[e]][0:ni] | EA[e][0:ew] ]
  const float* H; int ldh; int ni;
  const float* EA; int lde; int ew;
  const int* src; const int* dst;
  const float* hs; const float* hd; const float* ear;
  __device__ void init(int m, int M) {
    int mm = (m < M) ? m : (M - 1);
    hs  = H  + (long long)src[mm] * ldh;
    hd  = H  + (long long)dst[mm] * ldh;
    ear = EA + (long long)mm * lde;
  }
  __device__ float get(int k) const {          // k < K: branch-free addr select
    const float* p = (k < ni) ? (hs + k)
                   : ((k < 2 * ni) ? (hd + (k - ni)) : (ear + (k - 2 * ni)));
    return *p;
  }
  __device__ float getz(int k, int K) const { return (k < K) ? get(k) : 0.f; }
};

template <class AL>
__global__ __launch_bounds__(GTHREADS)
void k_gemm(AL al, const float* __restrict__ Wm, const float* __restrict__ bias,
            int M, int K, int N, float* __restrict__ Y, int ldy,
            float* __restrict__ cs, float* __restrict__ css) {
  __shared__ float Wl[MAXK * 16];       // 16-column W slice, zero padded
  __shared__ float rs[16], rss[16];
  const int tid = threadIdx.x;
  const int n0  = blockIdx.y * 16;
  const int K4  = (K + 3) & ~3;

  for (int i = tid; i < K4 * 16; i += GTHREADS) {
    int k = i >> 4, nn = i & 15, col = n0 + nn;
    Wl[i] = (k < K && col < N) ? Wm[(long long)k * N + col] : 0.f;
  }
  if (tid < 16) { rs[tid] = 0.f; rss[tid] = 0.f; }
  __syncthreads();

  const int wave = tid >> 5, lane = tid & 31;
  const int g = lane >> 4, n = lane & 15;     // lane group / column-in-tile
  const int m0 = (blockIdx.x * GWAVES + wave) * 16;
  al.init(m0 + n, M);                          // A row carried by this lane

  v8f acc = {};
  const int Kmain = K & ~3;                    // full steps: all k < K
  int kk = 0;
#pragma unroll 2
  for (; kk < Kmain; kk += 4) {
    const int ka = kk + 2 * g;                 // ISA layout: k = j + 2*(lane>>4)
    v2f a, b;
    a.x = al.get(ka);
    a.y = al.get(ka + 1);
    b.x = Wl[ka * 16 + n];
    b.y = Wl[(ka + 1) * 16 + n];
    acc = __builtin_amdgcn_wmma_f32_16x16x4_f32(false, a, false, b,
                                                (short)0, acc, false, false);
  }
  if (kk < K4) {                               // single zero-padded tail step
    const int ka = kk + 2 * g;
    v2f a, b;
    a.x = al.getz(ka, K);
    a.y = al.getz(ka + 1, K);
    b.x = Wl[ka * 16 + n];
    b.y = Wl[(ka + 1) * 16 + n];
    acc = __builtin_amdgcn_wmma_f32_16x16x4_f32(false, a, false, b,
                                                (short)0, acc, false, false);
  }

  // Epilogue: +bias, store, per-column sum / sum-of-squares for BatchNorm.
  const int col = n0 + n;
  if (col < N) {
    const float bv = bias[col];
    float s = 0.f, ss = 0.f;
#pragma unroll
    for (int j = 0; j < 8; ++j) {              // C/D: row = m0 + j + 8*g
      int row = m0 + j + 8 * g;
      if (row < M) {
        float v = acc[j] + bv;
        Y[(long long)row * ldy + col] = v;
        s += v; ss += v * v;
      }
    }
    atomicAdd(&rs[n], s);
    atomicAdd(&rss[n], ss);
  }
  __syncthreads();
  if (tid < 16 && (n0 + tid) < N) {
    atomicAdd(&cs[n0 + tid],  rs[tid]);
    atomicAdd(&css[n0 + tid], rss[tid]);
  }
}

// ---------------- message scatter-add (aggr[dst] += [h[src] | ea]) ----------
__global__ void k_scatter(const float* __restrict__ H, int ldh, int ni,
                          const float* __restrict__ EA, int lde, int eo,
                          const int* __restrict__ src, const int* __restrict__ dst,
                          float* __restrict__ aggr, int lda) {
  const int Wd = ni + eo;
  const long long total = (long long)NEDGES * Wd;
  long long i  = (long long)blockIdx.x * blockDim.x + threadIdx.x;
  long long st = (long long)gridDim.x * blockDim.x;
  for (; i < total; i += st) {
    int e = (int)(i / Wd);
    int c = (int)(i - (long long)e * Wd);
    float v = (c < ni) ? H[(long long)src[e] * ldh + c]
                       : EA[(long long)e * lde + (c - ni)];
    atomicAdd(&aggr[(long long)dst[e] * lda + c], v);
  }
}

// ---------------- pooling ---------------------------------------------------
__global__ void k_pool_nodes(const float* __restrict__ H, int ldh,
                             const int* __restrict__ batch,
                             float* __restrict__ z, float* __restrict__ cnt) {
  const long long total = (long long)NNODES * 64;
  long long i  = (long long)blockIdx.x * blockDim.x + threadIdx.x;
  long long st = (long long)gridDim.x * blockDim.x;
  for (; i < total; i += st) {
    int nidx = (int)(i >> 6), c = (int)(i & 63);
    int gph = batch[nidx];
    atomicAdd(&z[gph * 96 + c], H[(long long)nidx * ldh + c]);
    if (c == 0) atomicAdd(&cnt[gph], 1.f);
  }
}

__global__ void k_pool_edges(const float* __restrict__ EA, int lde,
                             const int* __restrict__ batch, const int* __restrict__ src,
                             float* __restrict__ z, float* __restrict__ cnt) {
  const long long total = (long long)NEDGES * 32;
  long long i  = (long long)blockIdx.x * blockDim.x + threadIdx.x;
  long long st = (long long)gridDim.x * blockDim.x;
  for (; i < total; i += st) {
    int e = (int)(i >> 5), c = (int)(i & 31);
    int gph = batch[src[e]];
    atomicAdd(&z[gph * 96 + 64 + c], EA[(long long)e * lde + c]);
    if (c == 0) atomicAdd(&cnt[gph], 1.f);
  }
}

__global__ void k_pool_div(float* __restrict__ z,
                           const float* __restrict__ ncnt, const float* __restrict__ ecnt) {
  int i = blockIdx.x * blockDim.x + threadIdx.x;
  if (i < NGRAPHS * 96) {
    int gph = i / 96, c = i % 96;
    float cnt = (c < 64) ? ncnt[gph] : ecnt[gph];
    z[i] /= fmaxf(cnt, 1.f);
  }
}

// ---------------- host-side helpers ----------------------------------------
static void gemm_plain(hipStream_t st, const float* X, int ldx,
                       const float* W, const float* b,
                       int M, int K, int N, float* Y, int ldy,
                       float* cs, float* css) {
  k_zero<<<1, 256, 0, st>>>(cs, 256);   // cs & css are contiguous (128+128)
  APlain al; al.X = X; al.ldx = ldx; al.row = nullptr;
  dim3 grid((M + 16 * GWAVES - 1) / (16 * GWAVES), (N + 15) / 16);
  k_gemm<APlain><<<grid, GTHREADS, 0, st>>>(al, W, b, M, K, N, Y, ldy, cs, css);
}

static void gemm_edge(hipStream_t st, const float* H, int ldh, int ni,
                      const float* EA, int ew,            // edge feats packed: lde == ew
                      const int* src, const int* dst,
                      const float* W, const float* b,
                      int N, float* Y, float* cs, float* css) {
  k_zero<<<1, 256, 0, st>>>(cs, 256);
  AEdge al{};
  al.H = H; al.ldh = ldh; al.ni = ni;
  al.EA = EA; al.lde = ew; al.ew = ew;
  al.src = src; al.dst = dst;
  al.hs = nullptr; al.hd = nullptr; al.ear = nullptr;
  int K = 2 * ni + ew;
  dim3 grid((NEDGES + 16 * GWAVES - 1) / (16 * GWAVES), (N + 15) / 16);
  k_gemm<AEdge><<<grid, GTHREADS, 0, st>>>(al, W, b, NEDGES, K, N, Y, N, cs, css);
}

static void bn_act(hipStream_t st, float* Y, int M, int N,
                   const float* gamma, const float* beta,
                   float* cs, float* css, float* sc, float* sh, int act) {
  k_bn_finalize<<<1, 128, 0, st>>>(cs, css, gamma, beta, N, 1.f / (float)M, sc, sh);
  long long total = (long long)M * N;
  long long want = (total + 255) / 256;
  int blocks = (int)(want < 8192 ? want : 8192);
  k_apply<<<blocks, 256, 0, st>>>(Y, total, N, sc, sh, act);
}

// ---------------- entry -----------------------------------------------------
extern "C" void kernel_launch(void* const* d_in, const int* in_sizes, int n_in,
                              void* d_out, int out_size, void* d_ws, size_t ws_size,
                              hipStream_t stream) {
  (void)in_sizes; (void)n_in; (void)out_size; (void)ws_size;

  // Param leaves in make_params() insertion order (4 per BN block).
  const float* P[68];
  for (int i = 0; i < 68; ++i) P[i] = (const float*)d_in[i];
  const float* h_in  = (const float*)d_in[68];   // [50000,11]
  const float* ea_in = (const float*)d_in[69];   // [800000,4]
  const int*   eidx  = (const int*)d_in[70];     // [2,800000]
  const int*   batch = (const int*)d_in[71];     // [50000]
  const int* src = eidx;
  const int* dst = eidx + NEDGES;

  char* w = (char*)d_ws;
  auto take = [&](size_t bytes) -> float* {
    float* p = (float*)w;
    w += (bytes + 255) & ~(size_t)255;
    return p;
  };
  float* EA_A = take((size_t)NEDGES * 64 * 4);
  float* EA_B = take((size_t)NEDGES * 64 * 4);
  float* H1   = take((size_t)NNODES * 128 * 4);
  float* H2   = take((size_t)NNODES * 128 * 4);
  float* AG   = take((size_t)NNODES * 192 * 4);
  float* Z    = take((size_t)NGRAPHS * 96 * 4);
  float* Y0   = take((size_t)NGRAPHS * 64 * 4);
  float* Y1   = take((size_t)NGRAPHS * 32 * 4);
  float* Y2   = take((size_t)NGRAPHS * 16 * 4);
  float* CS   = take(128 * 4);     // NOTE: CS/CSS contiguous (512B each slot)
  float* CSS  = take(128 * 4);
  float* SC   = take(128 * 4);
  float* SH   = take(128 * 4);
  float* NCNT = take(NGRAPHS * 4);
  float* ECNT = take(NGRAPHS * 4);

  // --- node preprocessing: 11 -> 22 -> 11 ---
  gemm_plain(stream, h_in, 11, P[0], P[1], NNODES, 11, 22, H2, 22, CS, CSS);
  bn_act(stream, H2, NNODES, 22, P[2], P[3], CS, CSS, SC, SH, 1);
  gemm_plain(stream, H2, 22, P[4], P[5], NNODES, 22, 11, H1, 11, CS, CSS);
  bn_act(stream, H1, NNODES, 11, P[6], P[7], CS, CSS, SC, SH, 1);

  // --- edge preprocessing: 4 -> 8 -> 4 ---
  gemm_plain(stream, ea_in, 4, P[8], P[9], NEDGES, 4, 8, EA_B, 8, CS, CSS);
  bn_act(stream, EA_B, NEDGES, 8, P[10], P[11], CS, CSS, SC, SH, 1);
  gemm_plain(stream, EA_B, 8, P[12], P[13], NEDGES, 8, 4, EA_A, 4, CS, CSS);
  bn_act(stream, EA_A, NEDGES, 4, P[14], P[15], CS, CSS, SC, SH, 1);

  // --- GNN layers ---
  const int cfg[5][4] = {{11,128,4,64},{128,128,64,64},{128,64,64,32},{64,64,32,32},{64,64,32,32}};
  float* hCur = H1;  float* hNext = H2;  int hw = 11;
  float* eaCur = EA_A; float* eaNext = EA_B; int ew = 4;

  for (int i = 0; i < 5; ++i) {
    const int ni = cfg[i][0], no = cfg[i][1], eo = cfg[i][3];
    const float* eW = P[16 + i * 8 + 0]; const float* eb = P[16 + i * 8 + 1];
    const float* eg = P[16 + i * 8 + 2]; const float* eB = P[16 + i * 8 + 3];
    const float* nW = P[16 + i * 8 + 4]; const float* nb = P[16 + i * 8 + 5];
    const float* ng = P[16 + i * 8 + 6]; const float* nB = P[16 + i * 8 + 7];

    // edge MLP on gathered [h_src | h_dst | ea]  (fused gather in GEMM)
    gemm_edge(stream, hCur, hw, ni, eaCur, ew, src, dst, eW, eb, eo, eaNext, CS, CSS);
    bn_act(stream, eaNext, NEDGES, eo, eg, eB, CS, CSS, SC, SH, 1);
    { float* t = eaCur; eaCur = eaNext; eaNext = t; }
    ew = eo;

    // aggregate messages [h_src | ea_new] at dst
    const int Wa = ni + eo;
    k_zero<<<4096, 256, 0, stream>>>(AG, (long long)NNODES * Wa);
    k_scatter<<<16384, 256, 0, stream>>>(hCur, hw, ni, eaCur, ew, eo, src, dst, AG, Wa);

    // node MLP
    gemm_plain(stream, AG, Wa, nW, nb, NNODES, Wa, no, hNext, no, CS, CSS);
    bn_act(stream, hNext, NNODES, no, ng, nB, CS, CSS, SC, SH, 1);
    { float* t = hCur; hCur = hNext; hNext = t; }
    hw = no;
  }

  // --- pooling: z = [seg_mean(h,batch) | seg_mean(ea,batch[src])] -> [1024,96]
  k_zero<<<64, 256, 0, stream>>>(Z, (long long)NGRAPHS * 96);
  k_zero<<<4, 256, 0, stream>>>(NCNT, NGRAPHS);
  k_zero<<<4, 256, 0, stream>>>(ECNT, NGRAPHS);
  k_pool_nodes<<<4096, 256, 0, stream>>>(hCur, hw, batch, Z, NCNT);
  k_pool_edges<<<8192, 256, 0, stream>>>(eaCur, ew, batch, src, Z, ECNT);
  k_pool_div<<<(NGRAPHS * 96 + 255) / 256, 256, 0, stream>>>(Z, NCNT, ECNT);

  // --- readout MLP ---
  gemm_plain(stream, Z, 96, P[56], P[57], NGRAPHS, 96, 64, Y0, 64, CS, CSS);
  bn_act(stream, Y0, NGRAPHS, 64, P[58], P[59], CS, CSS, SC, SH, 1);
  gemm_plain(stream, Y0, 64, P[60], P[61], NGRAPHS, 64, 32, Y1, 32, CS, CSS);
  bn_act(stream, Y1, NGRAPHS, 32, P[62], P[63], CS, CSS, SC, SH, 2);   // tanh
  gemm_plain(stream, Y1, 32, P[64], P[65], NGRAPHS, 32, 16, Y2, 16, CS, CSS);
  k_apply<<<64, 256, 0, stream>>>(Y2, (long long)NGRAPHS * 16, 16, nullptr, nullptr, 1); // ReLU, no BN
  gemm_plain(stream, Y2, 16, P[66], P[67], NGRAPHS, 16, 1, (float*)d_out, 1, CS, CSS);
}